// HetSAGPooling_33088428048492
// MI455X (gfx1250) — compile-verified
//
#include <hip/hip_runtime.h>
#include <math.h>

#define N_NODES 65536
#define N_EDGES 1048576
#define DIM     512
#define NBATCH  64

typedef __attribute__((ext_vector_type(2))) float v2f;
typedef __attribute__((ext_vector_type(8))) float v8f;

// ---------- helpers ----------

__device__ __forceinline__ float wave_sum(float v) {
#pragma unroll
    for (int off = 16; off > 0; off >>= 1) v += __shfl_xor(v, off, 32);
    return v;
}

// Correct float atomic-max via sign-split (dest must be initialized to -inf).
__device__ __forceinline__ void atomic_max_float(float* addr, float val) {
    if (val >= 0.0f) atomicMax((int*)addr, __float_as_int(val));
    else             atomicMin((unsigned int*)addr, __float_as_uint(val));
}

// ---------- init ----------

__global__ void fill_kernel(float* __restrict__ p, float val, int n) {
    int i = blockIdx.x * blockDim.x + threadIdx.x;
    if (i < n) p[i] = val;
}

// cat is 64 x 1024: cols [0,512) = running segment-sum (0), cols [512,1024) = running segment-max (-inf)
__global__ void cat_init_kernel(float* __restrict__ cat) {
    int i = blockIdx.x * blockDim.x + threadIdx.x;   // 64*1024 threads
    int col = i & 1023;
    cat[i] = (col < DIM) ? 0.0f : -INFINITY;
}

// ---------- LayerNorm + kqv projection (one wave32 per node) ----------

__global__ void ln_kqv_kernel(const float* __restrict__ x,
                              const float* __restrict__ g,
                              const float* __restrict__ bta,
                              const float* __restrict__ W,     // D x 3
                              const float* __restrict__ bkqv,  // 3
                              float* __restrict__ kk, float* __restrict__ qq,
                              float* __restrict__ vv,
                              float* __restrict__ meanA, float* __restrict__ rstdA) {
    int wave = threadIdx.x >> 5;
    int lane = threadIdx.x & 31;
    int node = blockIdx.x * (blockDim.x >> 5) + wave;
    if (node >= N_NODES) return;

    const float* xr = x + (size_t)node * DIM + lane * 16;
    float xv[16];
#pragma unroll
    for (int j = 0; j < 4; ++j) {                     // 4x global_load_b128 per lane
        float4 t = ((const float4*)xr)[j];
        xv[4*j+0] = t.x; xv[4*j+1] = t.y; xv[4*j+2] = t.z; xv[4*j+3] = t.w;
    }

    float s = 0.0f, s2 = 0.0f;
#pragma unroll
    for (int j = 0; j < 16; ++j) { s += xv[j]; s2 += xv[j] * xv[j]; }
    s  = wave_sum(s);
    s2 = wave_sum(s2);
    float mean = s * (1.0f / DIM);
    float var  = s2 * (1.0f / DIM) - mean * mean;
    float rstd = rsqrtf(var + 1e-5f);

    float a0 = 0.0f, a1 = 0.0f, a2 = 0.0f;
    int dbase = lane * 16;
#pragma unroll
    for (int j = 0; j < 16; ++j) {
        int d = dbase + j;
        float xn = (xv[j] - mean) * rstd * g[d] + bta[d];
        a0 += xn * W[d * 3 + 0];
        a1 += xn * W[d * 3 + 1];
        a2 += xn * W[d * 3 + 2];
    }
    a0 = wave_sum(a0); a1 = wave_sum(a1); a2 = wave_sum(a2);
    if (lane == 0) {
        kk[node]    = a0 + bkqv[0];
        qq[node]    = a1 + bkqv[1];
        vv[node]    = a2 + bkqv[2];
        meanA[node] = mean;
        rstdA[node] = rstd;
    }
}

// ---------- edge passes ----------

__global__ void edge_max_kernel(const int* __restrict__ ei,
                                const float* __restrict__ ksrc,
                                const float* __restrict__ qdst,
                                const float* __restrict__ krw,
                                const float* __restrict__ krb,
                                const float* __restrict__ prel, int rel,
                                float* __restrict__ m) {
    float kw = krw[rel], kb = krb[rel], pw = prel[rel];
    int stride = blockDim.x * gridDim.x;
    for (int e = blockIdx.x * blockDim.x + threadIdx.x; e < N_EDGES; e += stride) {
        if (e + stride < N_EDGES) {                    // gfx1250 global_prefetch_b8
            __builtin_prefetch(&ei[e + stride], 0, 1);
            __builtin_prefetch(&ei[N_EDGES + e + stride], 0, 1);
        }
        int src = ei[e];
        int dst = ei[N_EDGES + e];
        float logit = qdst[dst] * (ksrc[src] * kw + kb) * pw;
        atomic_max_float(&m[dst], logit);
    }
}

__global__ void edge_sum_kernel(const int* __restrict__ ei,
                                const float* __restrict__ ksrc,
                                const float* __restrict__ vsrc,
                                const float* __restrict__ qdst,
                                const float* __restrict__ krw,
                                const float* __restrict__ krb,
                                const float* __restrict__ vrw,
                                const float* __restrict__ vrb,
                                const float* __restrict__ prel, int rel,
                                const float* __restrict__ m,
                                float* __restrict__ den,
                                float* __restrict__ num) {
    float kw = krw[rel], kb = krb[rel], pw = prel[rel];
    float vw = vrw[rel], vb = vrb[rel];
    int stride = blockDim.x * gridDim.x;
    for (int e = blockIdx.x * blockDim.x + threadIdx.x; e < N_EDGES; e += stride) {
        if (e + stride < N_EDGES) {
            __builtin_prefetch(&ei[e + stride], 0, 1);
            __builtin_prefetch(&ei[N_EDGES + e + stride], 0, 1);
        }
        int src = ei[e];
        int dst = ei[N_EDGES + e];
        float logit = qdst[dst] * (ksrc[src] * kw + kb) * pw;
        float ex = __expf(logit - m[dst]);
        float vt = vsrc[src] * vw + vb;
        atomicAdd(&den[dst], ex);
        atomicAdd(&num[dst], ex * vt);
    }
}

// ---------- per-node score + batch-softmax max ----------

__global__ void score_max_kernel(const float* __restrict__ num,
                                 const float* __restrict__ den,
                                 const float* __restrict__ wout,
                                 const float* __restrict__ bout,
                                 const int* __restrict__ batch,
                                 float* __restrict__ score,
                                 float* __restrict__ pm) {
    int i = blockIdx.x * blockDim.x + threadIdx.x;
    if (i >= N_NODES) return;
    float d = den[i];
    float aggr = (d > 0.0f) ? (num[i] / d) : 0.0f;    // empty segments -> 0 (matches segment_sum)
    float z = aggr * wout[0] + bout[0];
    float sc = 0.5f * z * (1.0f + erff(z * 0.70710678118654752f));  // exact GELU
    score[i] = sc;
    atomic_max_float(&pm[batch[i]], sc);
}

__global__ void score_exp_kernel(const int* __restrict__ batch,
                                 const float* __restrict__ pm,
                                 float* __restrict__ score,   // in: score, out: exp(score - max)
                                 float* __restrict__ ps) {
    int i = blockIdx.x * blockDim.x + threadIdx.x;
    if (i >= N_NODES) return;
    int b = batch[i];
    float ex = __expf(score[i] - pm[b]);
    atomicAdd(&ps[b], ex);
    score[i] = ex;
}

// ---------- pooling accumulation: xw = xn * softmax_weight; segment add + max ----------

__global__ void pool_accum_kernel(const float* __restrict__ x,
                                  const float* __restrict__ g,
                                  const float* __restrict__ bta,
                                  const float* __restrict__ meanA,
                                  const float* __restrict__ rstdA,
                                  const float* __restrict__ score_e,
                                  const float* __restrict__ ps,
                                  const int* __restrict__ batch,
                                  float* __restrict__ cat) {     // 64 x 1024
    int node = blockIdx.x;
    int b = batch[node];
    float w    = score_e[node] / ps[b];
    float mean = meanA[node];
    float rstd = rstdA[node];
    const float* xr = x + (size_t)node * DIM;
    float* crow = cat + b * 1024;
    for (int d = threadIdx.x; d < DIM; d += blockDim.x) {
        float xn = (xr[d] - mean) * rstd * g[d] + bta[d];
        float xw = xn * w;
        atomicAdd(&crow[d], xw);
        atomic_max_float(&crow[DIM + d], xw);
    }
}

// ---------- final MLP: cat(64x1024) @ W(1024x512) + bias, via V_WMMA_F32_16X16X4_F32 ----------
// One wave per 16x16 output tile. Fragment layouts per CDNA5 ISA 7.12.2:
//   A 16x4 f32 : lanes 0-15 -> M rows, v0=K0,v1=K1 ; lanes 16-31 -> v0=K2,v1=K3
//   B 4x16 f32 : lanes 0-15 -> N cols, v0=K0,v1=K1 ; lanes 16-31 -> v0=K2,v1=K3
//   C 16x16 f32: vgpr r: lanes 0-15 M=r, lanes 16-31 M=r+8; N=lane%16

__global__ void pool_mlp_wmma_kernel(const float* __restrict__ cat,   // 64 x 1024
                                     const float* __restrict__ W,    // 1024 x 512 row-major
                                     const float* __restrict__ bias, // 512
                                     float* __restrict__ out,        // 64 x 1024 interleaved
                                     int out_col_off) {
    int wave = threadIdx.x >> 5;
    int lane = threadIdx.x & 31;
    int tile = blockIdx.x * (blockDim.x >> 5) + wave;  // 0..127 (Mt=4 x Nt=32)
    int tm = (tile >> 5) << 4;
    int tn = (tile & 31) << 4;
    int hl  = lane >> 4;       // 0: K pair {0,1}, 1: K pair {2,3}
    int l16 = lane & 15;

    v8f acc = {};
    const float* arow = cat + (tm + l16) * 1024 + 2 * hl;
    const float* bcol = W + (2 * hl) * 512 + tn + l16;
#pragma unroll 4
    for (int k = 0; k < 1024; k += 4) {
        v2f a; a.x = arow[0];  a.y = arow[1];
        v2f b; b.x = bcol[0];  b.y = bcol[512];
        acc = __builtin_amdgcn_wmma_f32_16x16x4_f32(
            /*neg_a=*/false, a, /*neg_b=*/false, b,
            /*c_mod=*/(short)0, acc, /*reuse_a=*/false, /*reuse_b=*/false);
        arow += 4;
        bcol += 4 * 512;
    }
#pragma unroll
    for (int r = 0; r < 8; ++r) {
        int row = tm + r + 8 * hl;
        int col = tn + l16;
        out[row * 1024 + out_col_off + col] = acc[r] + bias[col];
    }
}

// ---------- launch ----------

extern "C" void kernel_launch(void* const* d_in, const int* in_sizes, int n_in,
                              void* d_out, int out_size, void* d_ws, size_t ws_size,
                              hipStream_t stream) {
    const float* x_inst  = (const float*)d_in[0];
    const float* x_net   = (const float*)d_in[1];
    const float* ln_g_i  = (const float*)d_in[2];
    const float* ln_b_i  = (const float*)d_in[3];
    const float* ln_g_n  = (const float*)d_in[4];
    const float* ln_b_n  = (const float*)d_in[5];
    const float* Wkqv_i  = (const float*)d_in[6];
    const float* bkqv_i  = (const float*)d_in[7];
    const float* Wkqv_n  = (const float*)d_in[8];
    const float* bkqv_n  = (const float*)d_in[9];
    const float* k_rel_w = (const float*)d_in[10];
    const float* k_rel_b = (const float*)d_in[11];
    const float* v_rel_w = (const float*)d_in[12];
    const float* v_rel_b = (const float*)d_in[13];
    const float* p_rel   = (const float*)d_in[14];
    const float* w_out_i = (const float*)d_in[15];
    const float* b_out_i = (const float*)d_in[16];
    const float* w_out_n = (const float*)d_in[17];
    const float* b_out_n = (const float*)d_in[18];
    const float* W_mlp_i = (const float*)d_in[19];
    const float* b_mlp_i = (const float*)d_in[20];
    const float* W_mlp_n = (const float*)d_in[21];
    const float* b_mlp_n = (const float*)d_in[22];
    const int*   ei_i2n  = (const int*)d_in[23];
    const int*   ei_n2i  = (const int*)d_in[24];
    const int*   batch_i = (const int*)d_in[25];
    const int*   batch_n = (const int*)d_in[26];
    float* out = (float*)d_out;

    // workspace layout (floats); total ~5.2 MB
    float* ws = (float*)d_ws;
    const size_t N = N_NODES;
    float* kI = ws;           float* qI = ws + 1*N;  float* vI = ws + 2*N;
    float* kN = ws + 3*N;     float* qN = ws + 4*N;  float* vN = ws + 5*N;
    float* meanI = ws + 6*N;  float* rstdI = ws + 7*N;
    float* meanN = ws + 8*N;  float* rstdN = ws + 9*N;
    float* mNet  = ws + 10*N; float* sNet  = ws + 11*N; float* numNet  = ws + 12*N;
    float* mInst = ws + 13*N; float* sInst = ws + 14*N; float* numInst = ws + 15*N;
    float* scoreI = ws + 16*N;
    float* scoreN = ws + 17*N;
    float* pmI = ws + 18*N;        // 64
    float* pmN = pmI + NBATCH;     // 64
    float* psI = pmN + NBATCH;     // 64
    float* psN = psI + NBATCH;     // 64
    float* catI = psN + NBATCH;    // 64*1024
    float* catN = catI + 64*1024;  // 64*1024

    const int T = 256;
    const int nodeBlocks = (N_NODES + T - 1) / T;

    // --- init scratch ---
    fill_kernel<<<nodeBlocks, T, 0, stream>>>(mNet,  -INFINITY, N_NODES);
    fill_kernel<<<2*nodeBlocks, T, 0, stream>>>(sNet,  0.0f, 2*N_NODES);   // sNet + numNet
    fill_kernel<<<nodeBlocks, T, 0, stream>>>(mInst, -INFINITY, N_NODES);
    fill_kernel<<<2*nodeBlocks, T, 0, stream>>>(sInst, 0.0f, 2*N_NODES);   // sInst + numInst
    fill_kernel<<<1, T, 0, stream>>>(pmI, -INFINITY, 2*NBATCH);            // pmI + pmN
    fill_kernel<<<1, T, 0, stream>>>(psI, 0.0f, 2*NBATCH);                 // psI + psN
    cat_init_kernel<<<(64*1024)/T, T, 0, stream>>>(catI);
    cat_init_kernel<<<(64*1024)/T, T, 0, stream>>>(catN);

    // --- LayerNorm + kqv (8 waves/block, 1 node/wave) ---
    ln_kqv_kernel<<<N_NODES/8, T, 0, stream>>>(x_inst, ln_g_i, ln_b_i, Wkqv_i, bkqv_i,
                                               kI, qI, vI, meanI, rstdI);
    ln_kqv_kernel<<<N_NODES/8, T, 0, stream>>>(x_net, ln_g_n, ln_b_n, Wkqv_n, bkqv_n,
                                               kN, qN, vN, meanN, rstdN);

    // --- attention: i2n (rel 0, src=inst, dst=net), n2i (rel 1, src=net, dst=inst) ---
    const int EB = 1024;
    edge_max_kernel<<<EB, T, 0, stream>>>(ei_i2n, kI, qN, k_rel_w, k_rel_b, p_rel, 0, mNet);
    edge_max_kernel<<<EB, T, 0, stream>>>(ei_n2i, kN, qI, k_rel_w, k_rel_b, p_rel, 1, mInst);
    edge_sum_kernel<<<EB, T, 0, stream>>>(ei_i2n, kI, vI, qN, k_rel_w, k_rel_b,
                                          v_rel_w, v_rel_b, p_rel, 0, mNet, sNet, numNet);
    edge_sum_kernel<<<EB, T, 0, stream>>>(ei_n2i, kN, vN, qI, k_rel_w, k_rel_b,
                                          v_rel_w, v_rel_b, p_rel, 1, mInst, sInst, numInst);

    // --- scores + batch softmax ---
    score_max_kernel<<<nodeBlocks, T, 0, stream>>>(numInst, sInst, w_out_i, b_out_i,
                                                   batch_i, scoreI, pmI);
    score_max_kernel<<<nodeBlocks, T, 0, stream>>>(numNet,  sNet,  w_out_n, b_out_n,
                                                   batch_n, scoreN, pmN);
    score_exp_kernel<<<nodeBlocks, T, 0, stream>>>(batch_i, pmI, scoreI, psI);
    score_exp_kernel<<<nodeBlocks, T, 0, stream>>>(batch_n, pmN, scoreN, psN);

    // --- weighted pooling into cat = [segsum | segmax] ---
    pool_accum_kernel<<<N_NODES, T, 0, stream>>>(x_inst, ln_g_i, ln_b_i, meanI, rstdI,
                                                 scoreI, psI, batch_i, catI);
    pool_accum_kernel<<<N_NODES, T, 0, stream>>>(x_net,  ln_g_n, ln_b_n, meanN, rstdN,
                                                 scoreN, psN, batch_n, catN);

    // --- final MLP via WMMA: 128 tiles per type, 4 waves/block ---
    pool_mlp_wmma_kernel<<<32, 128, 0, stream>>>(catI, W_mlp_i, b_mlp_i, out, 0);
    pool_mlp_wmma_kernel<<<32, 128, 0, stream>>>(catN, W_mlp_n, b_mlp_n, out, DIM);
}